// KNNLoss_75179107549284
// MI455X (gfx1250) — compile-verified
//
#include <hip/hip_runtime.h>
#include <hip/hip_bf16.h>
#include <math.h>

typedef float v2f __attribute__((ext_vector_type(2)));
typedef float v8f __attribute__((ext_vector_type(8)));

#define DIM 128
#define KTOP 16
#define NEG_INF_F (-1.0e30f)

// Branch-free insert into ascending sorted 16-list (lst[0] is the current min).
__device__ __forceinline__ void topk_insert(float (&lst)[KTOP], float v) {
  if (v > lst[0]) {
    float cur = v;
#pragma unroll
    for (int t = 0; t < KTOP - 1; ++t) {
      float a  = lst[t + 1];
      float lo = fminf(cur, a);
      float hi = fmaxf(cur, a);
      lst[t] = lo;
      cur    = hi;
    }
    lst[KTOP - 1] = cur;
  }
}

// rn[i] = 1 / ||x_i||_2  ; one wave per row
__global__ __launch_bounds__(256) void knn_norm_kernel(
    const float* __restrict__ x, float* __restrict__ rn, int N) {
  const int wavesPerBlock = 8;
  const int row  = blockIdx.x * wavesPerBlock + (threadIdx.x >> 5);
  const int lane = threadIdx.x & 31;
  if (row >= N) return;  // wave-uniform
  const float* xr = x + (size_t)row * DIM;
  float s = 0.0f;
#pragma unroll
  for (int c = 0; c < DIM / 32; ++c) {
    float v = xr[lane + 32 * c];
    s += v * v;
  }
#pragma unroll
  for (int off = 16; off > 0; off >>= 1) s += __shfl_xor(s, off, 32);
  if (lane == 0) rn[row] = 1.0f / sqrtf(s);
}

// One block = one 16-row block of S. 4 waves split the N/16 column tiles.
// Per tile: 32x V_WMMA_F32_16X16X4_F32 over K=128, fused rn_i*rn_j scaling,
// streaming sum-exp (|S|<=1 so no max shift needed) and threshold top-16.
__global__ __launch_bounds__(128) void knn_tile_kernel(
    const float* __restrict__ x, const float* __restrict__ rn,
    const int* __restrict__ kp, float* __restrict__ partials, int N) {
  const int NT     = N >> 4;
  const int w      = threadIdx.x >> 5;
  const int lane   = threadIdx.x & 31;
  const int lane15 = lane & 15;
  const int hi     = lane >> 4;
  const int rowBase = blockIdx.x << 4;

  __shared__ float tileLds[4][16][17];   // per-wave transpose staging
  __shared__ float sumLds[4][32][8];     // per-lane sum-exp partials
  __shared__ float listLds[4][32][16];   // per-lane top-16 candidate lists
  __shared__ float termLds[16];          // per-row loss terms

  // A panel (16 rows x 128) register-resident: lane15 = M, hi selects K pair.
  v2f aReg[32];
  const float* aRow = x + (size_t)(rowBase + lane15) * DIM + 2 * hi;
#pragma unroll
  for (int kc = 0; kc < 32; ++kc)
    aReg[kc] = *reinterpret_cast<const v2f*>(aRow + kc * 4);

  float rnA[8];
#pragma unroll
  for (int v = 0; v < 8; ++v) rnA[v] = rn[rowBase + v + 8 * hi];

  float rowSum[8];
#pragma unroll
  for (int v = 0; v < 8; ++v) rowSum[v] = 0.0f;

  float lst[KTOP];
#pragma unroll
  for (int t = 0; t < KTOP; ++t) lst[t] = NEG_INF_F;

  const int r = lane >> 1;  // top-k: this lane owns row r, column half h
  const int h = lane & 1;
  const int nIter = (NT + 3) >> 2;

  for (int jj = 0; jj < nIter; ++jj) {
    const int j = (jj << 2) + w;     // wave-interleaved column tiles
    const bool act = (j < NT);       // wave-uniform predicate
    if (act) {
      const int colBase = j << 4;
      const float* bRow = x + (size_t)(colBase + lane15) * DIM + 2 * hi;
      v8f c = {};
#pragma unroll
      for (int kc = 0; kc < 32; ++kc) {
        v2f b = *reinterpret_cast<const v2f*>(bRow + kc * 4);
        c = __builtin_amdgcn_wmma_f32_16x16x4_f32(
            /*neg_a=*/false, aReg[kc], /*neg_b=*/false, b,
            /*c_mod=*/(short)0, c, /*reuse_a=*/false, /*reuse_b=*/false);
      }
      const float rnB = rn[colBase + lane15];
#pragma unroll
      for (int v = 0; v < 8; ++v) {
        float sv = c[v] * rnA[v] * rnB;        // S[rowBase+v+8hi][colBase+lane15]
        rowSum[v] += __expf(sv);               // denominator includes diagonal
        tileLds[w][v + 8 * hi][lane15] = sv;
      }
    }
    __syncthreads();
    if (act) {
      const int colBase = j << 4;
      const int rowG = rowBase + r;
#pragma unroll
      for (int cc = 0; cc < 8; ++cc) {
        const int colG = colBase + 8 * h + cc;
        float sv = tileLds[w][r][8 * h + cc];
        if (colG != rowG) topk_insert(lst, sv);  // top-k excludes diagonal
      }
    }
    __syncthreads();
  }

  // Publish per-lane partials, merge in fixed order (deterministic).
#pragma unroll
  for (int v = 0; v < 8; ++v) sumLds[w][lane][v] = rowSum[v];
#pragma unroll
  for (int t = 0; t < KTOP; ++t) listLds[w][lane][t] = lst[t];
  __syncthreads();

  if (threadIdx.x < 16) {
    const int rr    = threadIdx.x;
    const int hiSel = (rr >> 3) << 4;
    const int vSel  = rr & 7;
    float se = 0.0f;
    for (int w2 = 0; w2 < 4; ++w2)
#pragma unroll
      for (int l = 0; l < 16; ++l) se += sumLds[w2][hiSel + l][vSel];

    float fl[KTOP];
#pragma unroll
    for (int t = 0; t < KTOP; ++t) fl[t] = NEG_INF_F;
    for (int w2 = 0; w2 < 4; ++w2)
      for (int hh = 0; hh < 2; ++hh)
#pragma unroll
        for (int t = 0; t < KTOP; ++t)
          topk_insert(fl, listLds[w2][2 * rr + hh][t]);

    float topSum = 0.0f;
#pragma unroll
    for (int t = 0; t < KTOP; ++t) topSum += fl[t];

    const float kF = (float)(*kp);
    termLds[rr] = kF * __logf(se) - topSum;  // loss contribution of row rr
  }
  __syncthreads();
  if (threadIdx.x == 0) {
    float s = 0.0f;
#pragma unroll
    for (int rr2 = 0; rr2 < 16; ++rr2) s += termLds[rr2];
    partials[blockIdx.x] = s;
  }
}

__global__ __launch_bounds__(256) void knn_reduce_kernel(
    const float* __restrict__ p, int n, float* __restrict__ out) {
  __shared__ float red[256];
  float s = 0.0f;
  for (int i = threadIdx.x; i < n; i += 256) s += p[i];
  red[threadIdx.x] = s;
  __syncthreads();
  for (int off = 128; off > 0; off >>= 1) {
    if ((int)threadIdx.x < off) red[threadIdx.x] += red[threadIdx.x + off];
    __syncthreads();
  }
  if (threadIdx.x == 0) out[0] = red[0];
}

extern "C" void kernel_launch(void* const* d_in, const int* in_sizes, int n_in,
                              void* d_out, int out_size, void* d_ws, size_t ws_size,
                              hipStream_t stream) {
  const float* x  = (const float*)d_in[0];
  const int*   kp = (const int*)d_in[1];
  const int N  = in_sizes[0] / DIM;   // 8192
  const int NT = N >> 4;              // 512 row/column tiles

  float* rn       = (float*)d_ws;     // N floats
  float* partials = rn + N;           // NT floats

  knn_norm_kernel<<<(N + 7) / 8, 256, 0, stream>>>(x, rn, N);
  knn_tile_kernel<<<NT, 128, 0, stream>>>(x, rn, kp, partials, N);
  knn_reduce_kernel<<<1, 256, 0, stream>>>(partials, NT, (float*)d_out);
}